// BCMSELoss_22694607192511
// MI455X (gfx1250) — compile-verified
//
#include <hip/hip_runtime.h>
#include <hip/hip_bf16.h>
#include <math.h>

// ---------------------------------------------------------------------------
// BCMSE loss: streaming reduction over (B,9) f32 outputs/targets -> 1 scalar.
// Memory-bound: ~302 MB read @ 23.3 TB/s => ~13us floor. Strategy:
//   kernel 1: grid-stride, 5 per-lane f32 accumulators, WMMA-based
//             deterministic wave reduction, LDS cross-wave reduce,
//             per-block partials to d_ws.
//   kernel 2: single wave32 reduces partials in fixed order (deterministic),
//             applies final formula.
// WMMA trick: V_WMMA_F32_16X16X4_F32 with B = ones(4x16):
//   D[m][n] = a0[m] + a1[m] + a0[m+16] + a1[m+16]   (for every column n)
// so one WMMA + 7 adds + one shfl_xor(16) = exact full-wave f32 sum.
// ---------------------------------------------------------------------------

typedef float v2f __attribute__((ext_vector_type(2)));
typedef float v8f __attribute__((ext_vector_type(8)));

#define NBLK 1024
#define NTHR 256
#define NQ   5   // extra, norm, p0, p1, p2

// Deterministic full-wave32 sum using v_wmma_f32_16x16x4_f32.
// Requires EXEC all-ones (call from convergent code only).
__device__ __forceinline__ float wave_sum_wmma(float s) {
  v2f a; a.x = s;    a.y = 0.0f;   // A[m][k]: lane m holds {K0,K1}, lane m+16 {K2,K3}
  v2f b; b.x = 1.0f; b.y = 1.0f;   // B = ones(4x16)
  v8f c = {};
  v8f d = __builtin_amdgcn_wmma_f32_16x16x4_f32(
      /*neg_a=*/false, a, /*neg_b=*/false, b,
      /*c_mod=*/(short)0, c, /*reuse_a=*/false, /*reuse_b=*/false);
  // lane n<16 now holds D[r][n]=rowsum R[r] (r=0..7); lane n+16 holds R[r+8]
  float t = d[0] + d[1] + d[2] + d[3] + d[4] + d[5] + d[6] + d[7];
  t += __shfl_xor(t, 16, 32);      // combine the two 8-row halves
  return t;                        // full 32-lane sum in every lane
}

__device__ __forceinline__ float angle_sq(float o, float t, float& extra) {
  float fo = floorf(o);
  extra += fabsf(fo);
  float om = o - fo;               // jnp.remainder(o, 1.0) in [0,1)
  float tt = t;
  if (fabsf(om - t) > 0.5f) tt = (t < om) ? t + 1.0f : t - 1.0f;
  float d = om - tt;
  return d * d;
}

__device__ __forceinline__ float pymod(float v, float n) {
  // jnp.remainder(v, n), n > 0: result in [0, n)
  return v - floorf(v / n) * n;
}

__global__ void __launch_bounds__(NTHR)
bcmse_partials(const float* __restrict__ outs, const float* __restrict__ tgts,
               long long B, float* __restrict__ parts) {
  float s_extra = 0.0f, s_norm = 0.0f, s_p0 = 0.0f, s_p1 = 0.0f, s_p2 = 0.0f;

  const long long stride = (long long)gridDim.x * blockDim.x;
  for (long long row = (long long)blockIdx.x * blockDim.x + threadIdx.x;
       row < B; row += stride) {
    const float* o = outs + row * 9;
    const float* t = tgts + row * 9;

    // prefetch next grid-stride row (emits global_prefetch_b8)
    if (row + stride < B) {
      __builtin_prefetch(outs + (row + stride) * 9, 0, 0);
      __builtin_prefetch(tgts + (row + stride) * 9, 0, 0);
    }

    float o0 = o[0], o1 = o[1], o2 = o[2], o3 = o[3], o4 = o[4];
    float o5 = o[5], o6 = o[6], o7 = o[7], o8 = o[8];
    float t0 = t[0], t1 = t[1], t2 = t[2], t3 = t[3], t4 = t[4];
    float t5 = t[5], t6 = t[6], t7 = t[7], t8 = t[8];

    // scalar columns {0,3}
    float d;
    d = o0 - t0; s_p0 += d * d;
    d = o3 - t3; s_p0 += d * d;

    // angle columns {1,2,4,5}
    s_p1 += angle_sq(o1, t1, s_extra);
    s_p1 += angle_sq(o2, t2, s_extra);
    s_p1 += angle_sq(o4, t4, s_extra);
    s_p1 += angle_sq(o5, t5, s_extra);

    // vector columns 6:9
    float n2 = o6 * o6 + o7 * o7 + o8 * o8;
    float nr = sqrtf(n2);
    s_norm += nr;
    d = pymod(o6, nr) - t6; s_p2 += d * d;
    d = pymod(o7, nr) - t7; s_p2 += d * d;
    d = pymod(o8, nr) - t8; s_p2 += d * d;
  }

  // --- deterministic wave reduction via WMMA (all threads convergent here) ---
  float w0 = wave_sum_wmma(s_extra);
  float w1 = wave_sum_wmma(s_norm);
  float w2 = wave_sum_wmma(s_p0);
  float w3 = wave_sum_wmma(s_p1);
  float w4 = wave_sum_wmma(s_p2);

  // --- cross-wave reduction in LDS (8 waves per block), fixed order ---
  __shared__ float wred[(NTHR / 32) * NQ];
  const int lane = threadIdx.x & 31;
  const int wave = threadIdx.x >> 5;
  if (lane == 0) {
    wred[wave * NQ + 0] = w0;
    wred[wave * NQ + 1] = w1;
    wred[wave * NQ + 2] = w2;
    wred[wave * NQ + 3] = w3;
    wred[wave * NQ + 4] = w4;
  }
  __syncthreads();
  if (threadIdx.x < NQ) {
    float acc = 0.0f;
    for (int w = 0; w < NTHR / 32; ++w) acc += wred[w * NQ + threadIdx.x];
    parts[(long long)blockIdx.x * NQ + threadIdx.x] = acc;
  }
}

__global__ void __launch_bounds__(32)
bcmse_finalize(const float* __restrict__ parts, int nparts, long long B,
               float* __restrict__ out) {
  const int lane = threadIdx.x;  // exactly one wave32
  float s[NQ] = {0.f, 0.f, 0.f, 0.f, 0.f};
  for (int i = lane; i < nparts; i += 32) {
#pragma unroll
    for (int q = 0; q < NQ; ++q) s[q] += parts[i * NQ + q];
  }
  float tot[NQ];
#pragma unroll
  for (int q = 0; q < NQ; ++q) tot[q] = wave_sum_wmma(s[q]);

  if (lane == 0) {
    const float extras = tot[0], norms = tot[1];
    const float p0 = tot[2], p1 = tot[3], p2 = tot[4];
    const float fB = (float)B;
    const float mse = (p0 + p1 + p2) / (9.0f * fB);
    const float c0 = extras / fB / 10.0f;   // CONSTANT_WEIGHT
    const float c1 = norms  / fB / 10.0f;
    float amount;
    if (p0 > p1 && p0 > p2)      amount = 0.0f;
    else if (p0 > p1 && p0 < p2) amount = c1;
    else if (p0 < p1 && p0 > p2) amount = c0;
    else                         amount = c0 + c1;
    out[0] = mse + amount;
  }
}

extern "C" void kernel_launch(void* const* d_in, const int* in_sizes, int n_in,
                              void* d_out, int out_size, void* d_ws, size_t ws_size,
                              hipStream_t stream) {
  (void)n_in; (void)out_size; (void)ws_size;
  const float* outs = (const float*)d_in[0];
  const float* tgts = (const float*)d_in[1];
  const long long B = (long long)in_sizes[0] / 9;

  float* parts = (float*)d_ws;         // NBLK * NQ floats = 20 KB
  float* out   = (float*)d_out;

  bcmse_partials<<<NBLK, NTHR, 0, stream>>>(outs, tgts, B, parts);
  bcmse_finalize<<<1, 32, 0, stream>>>(parts, NBLK, B, out);
}